// VariationalGCNEncoder_2473901162944
// MI455X (gfx1250) — compile-verified
//
#include <hip/hip_runtime.h>

#define N_NODES 50000
#define N_EDGES 800000
#define IN_CH   128
#define HID     128
#define OUT_CH  64
#define BN_EPS  1e-5f

typedef __attribute__((ext_vector_type(2))) float v2f;
typedef __attribute__((ext_vector_type(8))) float v8f;

// ---------------------------------------------------------------- utilities
__global__ void k_fill(float* __restrict__ p, float v, long n) {
  long i = (long)blockIdx.x * blockDim.x + threadIdx.x;
  long stride = (long)gridDim.x * blockDim.x;
  for (; i < n; i += stride) p[i] = v;
}

// ---------------------------------------------------------------- degrees / norms
__global__ void k_deg(const long long* __restrict__ dst, float* __restrict__ deg, int E) {
  int e = blockIdx.x * blockDim.x + threadIdx.x;
  if (e < E) atomicAdd(&deg[(int)dst[e]], 1.0f);
}

__global__ void k_norm(const long long* __restrict__ src, const long long* __restrict__ dst,
                       const float* __restrict__ deg, float* __restrict__ nrm, int E) {
  int e = blockIdx.x * blockDim.x + threadIdx.x;
  if (e < E) nrm[e] = rsqrtf(deg[(int)src[e]]) * rsqrtf(deg[(int)dst[e]]);
}

__global__ void k_selfnorm(const float* __restrict__ deg, float* __restrict__ sn, int n) {
  int i = blockIdx.x * blockDim.x + threadIdx.x;
  if (i < n) sn[i] = 1.0f / deg[i];   // dinv[i]*dinv[i]
}

// ---------------------------------------------------------------- WMMA f32 GEMM
// C[M,N] = A[M,K] @ B[K,N], N multiple of 64.
// One wave computes a 16x64 strip: 4 accumulator tiles along N. Per K-step of 4
// the A fragment is loaded ONCE and fed to 4 V_WMMA_F32_16X16X4_F32 ops
// (4x less A traffic than per-tile waves; A is the streaming operand).
// Layouts per CDNA5 ISA 7.12.2:
//   A 16x4:  lane&15 = M, (lane>>4)*2 = K base, 2 VGPRs hold K, K+1
//   B 4x16:  lane&15 = N, (lane>>4)*2 = K base, 2 VGPRs hold K, K+1
//   C 16x16: 8 VGPRs, row = vgpr + (lane>=16 ? 8 : 0), col = lane&15
__global__ void k_gemm_wmma_f32(const float* __restrict__ A, const float* __restrict__ B,
                                float* __restrict__ C, int M, int K, int N) {
  int wave = (int)((blockIdx.x * blockDim.x + threadIdx.x) >> 5);
  int lane = threadIdx.x & 31;
  int groupsN = N >> 6;                     // 64-wide strips
  int nWaves  = (M >> 4) * groupsN;
  if (wave >= nWaves) return;               // wave-uniform: EXEC all-1s for WMMA
  int tm = wave / groupsN;
  int tg = wave - tm * groupsN;
  int m  = (tm << 4) + (lane & 15);
  int n0 = (tg << 6) + (lane & 15);
  int kh = (lane >> 4) << 1;                // 0 or 2

  v8f acc0 = {}, acc1 = {}, acc2 = {}, acc3 = {};
  const float* Arow = A + (long)m * K + kh;
#pragma unroll 4
  for (int k = 0; k < K; k += 4) {
    v2f a = *(const v2f*)(Arow + k);                    // 8B aligned (kh even)
    const float* B0 = B + (long)(k + kh) * N + n0;
    const float* B1 = B + (long)(k + kh + 1) * N + n0;
    v2f b0, b1, b2, b3;
    b0.x = B0[0];  b0.y = B1[0];
    b1.x = B0[16]; b1.y = B1[16];
    b2.x = B0[32]; b2.y = B1[32];
    b3.x = B0[48]; b3.y = B1[48];
    acc0 = __builtin_amdgcn_wmma_f32_16x16x4_f32(false, a, false, b0, (short)0, acc0, false, false);
    acc1 = __builtin_amdgcn_wmma_f32_16x16x4_f32(false, a, false, b1, (short)0, acc1, false, false);
    acc2 = __builtin_amdgcn_wmma_f32_16x16x4_f32(false, a, false, b2, (short)0, acc2, false, false);
    acc3 = __builtin_amdgcn_wmma_f32_16x16x4_f32(false, a, false, b3, (short)0, acc3, false, false);
  }

  int rbase = (tm << 4) + ((lane >> 4) << 3);
#pragma unroll
  for (int r = 0; r < 8; ++r) {
    float* Crow = C + (long)(rbase + r) * N + n0;
    Crow[0]  = acc0[r];
    Crow[16] = acc1[r];
    Crow[32] = acc2[r];
    Crow[48] = acc3[r];
  }
}

// ---------------------------------------------------------------- edge scatter
// out[dst] += T[src] * norm  -- C/4 lanes per edge, float4 gather (L2-resident)
template <int C>
__global__ void k_scatter(const long long* __restrict__ src, const long long* __restrict__ dst,
                          const float* __restrict__ nrm, const float* __restrict__ T,
                          float* __restrict__ out, int E) {
  constexpr int LPE = C / 4;
  long tid = (long)blockIdx.x * blockDim.x + threadIdx.x;
  long e   = tid / LPE;
  int  c4  = (int)(tid - e * LPE) * 4;
  if (e >= E) return;
  int s = (int)src[e], d = (int)dst[e];
  float w = nrm[e];
  float4 v = *(const float4*)(T + (long)s * C + c4);
  float* o = out + (long)d * C + c4;
  atomicAdd(o + 0, v.x * w);
  atomicAdd(o + 1, v.y * w);
  atomicAdd(o + 2, v.z * w);
  atomicAdd(o + 3, v.w * w);
}

// out[i,c] += T[i,c]*selfnorm[i] + bias[c]
__global__ void k_selfloop_bias(const float* __restrict__ T, const float* __restrict__ sn,
                                const float* __restrict__ bias, float* __restrict__ out,
                                int N, int C) {
  long tid = (long)blockIdx.x * blockDim.x + threadIdx.x;
  if (tid >= (long)N * C) return;
  long i = tid / C;
  int  c = (int)(tid - i * C);
  out[tid] += T[tid] * sn[i] + bias[c];
}

// ---------------------------------------------------------------- batch norm
// stats[0..C)   = sum per channel,  stats[C..2C) = sumsq per channel
__global__ void k_bn_stats(const float* __restrict__ h, float* __restrict__ stats, int N, int C) {
  int c = threadIdx.x;                     // blockDim.x == C
  int row0 = blockIdx.x * 256;
  int row1 = row0 + 256; if (row1 > N) row1 = N;
  float s = 0.f, ss = 0.f;
  for (int i = row0; i < row1; ++i) {
    float v = h[(long)i * C + c];
    s += v; ss += v * v;
  }
  atomicAdd(&stats[c], s);
  atomicAdd(&stats[C + c], ss);
}

__global__ void k_bn_relu(float* __restrict__ h, const float* __restrict__ stats,
                          const float* __restrict__ gamma, const float* __restrict__ beta,
                          int N, int C) {
  long tid = (long)blockIdx.x * blockDim.x + threadIdx.x;
  if (tid >= (long)N * C) return;
  int c = (int)(tid % C);
  float inv_n = 1.0f / (float)N;
  float mean = stats[c] * inv_n;
  float var  = stats[C + c] * inv_n - mean * mean;
  float v = (h[tid] - mean) * rsqrtf(var + BN_EPS) * gamma[c] + beta[c];
  h[tid] = fmaxf(v, 0.0f);
}

// ---------------------------------------------------------------- launch
extern "C" void kernel_launch(void* const* d_in, const int* in_sizes, int n_in,
                              void* d_out, int out_size, void* d_ws, size_t ws_size,
                              hipStream_t stream) {
  (void)in_sizes; (void)n_in; (void)out_size; (void)ws_size;

  const float*     x    = (const float*)d_in[0];
  const long long* eidx = (const long long*)d_in[1];   // int64 per reference
  const long long* src  = eidx;
  const long long* dst  = eidx + N_EDGES;
  const float* W1  = (const float*)d_in[2];
  const float* b1  = (const float*)d_in[3];
  const float* g1  = (const float*)d_in[4];
  const float* be1 = (const float*)d_in[5];
  const float* W2  = (const float*)d_in[6];
  const float* b2  = (const float*)d_in[7];
  const float* g2  = (const float*)d_in[8];
  const float* be2 = (const float*)d_in[9];
  const float* Wmu = (const float*)d_in[10];
  const float* bmu = (const float*)d_in[11];
  const float* Wls = (const float*)d_in[12];
  const float* bls = (const float*)d_in[13];

  float* out_mu = (float*)d_out;
  float* out_ls = out_mu + (size_t)N_NODES * OUT_CH;

  float* ws    = (float*)d_ws;
  float* deg   = ws;  ws += N_NODES;
  float* sn    = ws;  ws += N_NODES;
  float* nrm   = ws;  ws += N_EDGES;
  float* T     = ws;  ws += (size_t)N_NODES * HID;    // pre-aggregation h@W
  float* H     = ws;  ws += (size_t)N_NODES * HID;    // aggregated / activated
  float* stats = ws;  ws += 2 * HID;

  const int B = 256;
  auto cdiv = [](long a, long b) { return (int)((a + b - 1) / b); };

  // -- normalization terms (deg starts at 1.0 => self-loop counted)
  k_fill<<<cdiv(N_NODES, B), B, 0, stream>>>(deg, 1.0f, N_NODES);
  k_deg<<<cdiv(N_EDGES, B), B, 0, stream>>>(dst, deg, N_EDGES);
  k_norm<<<cdiv(N_EDGES, B), B, 0, stream>>>(src, dst, deg, nrm, N_EDGES);
  k_selfnorm<<<cdiv(N_NODES, B), B, 0, stream>>>(deg, sn, N_NODES);

  // one GCN layer: out = scatter(in@W) + selfloop + bias   (out may alias in)
  auto gcn = [&](const float* in, int K, const float* W, const float* bias, int C, float* out) {
    int nWaves = (N_NODES / 16) * (C / 64);
    k_gemm_wmma_f32<<<cdiv((long)nWaves * 32, B), B, 0, stream>>>(in, W, T, N_NODES, K, C);
    k_fill<<<2048, B, 0, stream>>>(out, 0.0f, (long)N_NODES * C);
    long sthreads = (long)N_EDGES * (C / 4);
    if (C == 128)
      k_scatter<128><<<cdiv(sthreads, B), B, 0, stream>>>(src, dst, nrm, T, out, N_EDGES);
    else
      k_scatter<64><<<cdiv(sthreads, B), B, 0, stream>>>(src, dst, nrm, T, out, N_EDGES);
    k_selfloop_bias<<<cdiv((long)N_NODES * C, B), B, 0, stream>>>(T, sn, bias, out, N_NODES, C);
  };

  auto bn_relu = [&](float* h, const float* gamma, const float* beta) {
    k_fill<<<1, B, 0, stream>>>(stats, 0.0f, 2 * HID);
    k_bn_stats<<<cdiv(N_NODES, 256), HID, 0, stream>>>(h, stats, N_NODES, HID);
    k_bn_relu<<<cdiv((long)N_NODES * HID, B), B, 0, stream>>>(h, stats, gamma, beta, N_NODES, HID);
  };

  // layer 1: x -> H
  gcn(x, IN_CH, W1, b1, HID, H);
  bn_relu(H, g1, be1);
  // layer 2: H -> H  (gemm reads H into T before H is zeroed; stream-ordered)
  gcn(H, HID, W2, b2, HID, H);
  bn_relu(H, g2, be2);
  // heads
  gcn(H, HID, Wmu, bmu, OUT_CH, out_mu);
  gcn(H, HID, Wls, bls, OUT_CH, out_ls);
}